// HGT_49349174231107
// MI455X (gfx1250) — compile-verified
//
#include <hip/hip_runtime.h>
#include <math.h>

// ---------------------------------------------------------------------------
// HGT (3 layers) + GAT scorer + SAGPool top-K + MLP for gfx1250 (MI455X).
// Dense GEMMs use V_WMMA_F32_16X16X4_F32 (full-precision f32 matrix path),
// one wave per 16x(16*NT) output strip with A-fragment reuse across NT tiles.
// Scatter-softmax uses ordered-uint atomicMax + atomicAdd with deferred
// segment normalization.
// ---------------------------------------------------------------------------

typedef float v2f __attribute__((ext_vector_type(2)));
typedef float v8f __attribute__((ext_vector_type(8)));

#define HID 128
#define HEADS 4
#define DIM 32
#define KQV 384
#define TOPK 8

static __device__ __forceinline__ float gelu_exact(float x) {
    return 0.5f * x * (1.0f + erff(x * 0.70710678118654752440f));
}

// order-preserving float <-> uint encoding for atomicMax on floats
static __device__ __forceinline__ unsigned int encF(float f) {
    unsigned int u = __float_as_uint(f);
    return (u & 0x80000000u) ? ~u : (u | 0x80000000u);
}
static __device__ __forceinline__ float decF(unsigned int u) {
    return (u & 0x80000000u) ? __uint_as_float(u & 0x7FFFFFFFu)
                             : __uint_as_float(~u);
}
#define ENC_NEG_INF 0x007FFFFFu  // encF(-inf)

// ---------------------------------------------------------------------------
// f32 WMMA GEMM: C[M,N] = A[M,K] @ B[K,N] (+bias) (+gated skip).
// One wave computes a 16 x (16*NT) strip: the A fragment (one b64 load per
// K-step) is reused across NT WMMAs. M mult of 16, N mult of 16*NT, K mult 4.
// epilogue: 0 -> C = acc + bias
//           1 -> s = sigmoid(*gate); C = s*(acc+bias) + (1-s)*skipX
// ---------------------------------------------------------------------------
template <int NT>
__global__ __launch_bounds__(128) void wmma_gemm_f32_kernel(
    const float* __restrict__ A, int lda, const float* __restrict__ B, int ldb,
    const float* __restrict__ bias, float* __restrict__ C, int ldc, int M,
    int N, int K, int epilogue, const float* __restrict__ skipX, int ldskip,
    const float* __restrict__ gatePtr) {
    const int wavesPerBlock = blockDim.x >> 5;
    const int waveId = blockIdx.x * wavesPerBlock + (threadIdx.x >> 5);
    const int lane = threadIdx.x & 31;
    const int nstrips = N / (16 * NT);
    const int mtiles = M >> 4;
    if (waveId >= nstrips * mtiles) return;  // whole-wave exit: EXEC all-1s
    const int tm = waveId / nstrips;
    const int ts = waveId % nstrips;
    const int r = lane & 15;   // A row-within-tile / B col-within-tile
    const int g = lane >> 4;   // lane group selects K pair / row half

    const float* Ap = A + (size_t)(tm * 16 + r) * lda;   // this lane's A row
    const float* Bp = B + ts * (16 * NT) + r;            // this lane's B col

    v8f acc[NT];
#pragma unroll
    for (int j = 0; j < NT; ++j) acc[j] = (v8f){0.f, 0.f, 0.f, 0.f,
                                                0.f, 0.f, 0.f, 0.f};

    for (int k0 = 0; k0 < K; k0 += 4) {
        const int ka = k0 + 2 * g;
        const v2f a = *(const v2f*)(Ap + ka);  // 8B aligned: lda, ka even
#pragma unroll
        for (int j = 0; j < NT; ++j) {
            v2f b;
            b.x = Bp[(size_t)ka * ldb + j * 16];
            b.y = Bp[(size_t)(ka + 1) * ldb + j * 16];
            acc[j] = __builtin_amdgcn_wmma_f32_16x16x4_f32(
                false, a, false, b, (short)0, acc[j], false, false);
        }
    }

    float gate = 1.0f, omg = 0.0f;
    if (epilogue == 1) {
        const float sv = 1.0f / (1.0f + expf(-gatePtr[0]));
        gate = sv;
        omg = 1.0f - sv;
    }
#pragma unroll
    for (int j = 0; j < NT; ++j) {
        const int col = ts * (16 * NT) + j * 16 + r;
        const float bv = bias ? bias[col] : 0.0f;
#pragma unroll
        for (int v = 0; v < 8; ++v) {
            const int row = tm * 16 + v + 8 * g;  // C/D layout: M=v (+8 hi)
            float val = acc[j][v] + bv;
            if (epilogue == 1)
                val = gate * val + omg * skipX[(size_t)row * ldskip + col];
            C[(size_t)row * ldc + col] = val;
        }
    }
}

// ---------------------------------------------------------------------------
// Fills
// ---------------------------------------------------------------------------
__global__ void fill_f32_kernel(float* p, float v, int n) {
    int t = blockIdx.x * blockDim.x + threadIdx.x;
    if (t < n) p[t] = v;
}
__global__ void fill_u32_kernel(unsigned int* p, unsigned int v, int n) {
    int t = blockIdx.x * blockDim.x + threadIdx.x;
    if (t < n) p[t] = v;
}

// ---------------------------------------------------------------------------
// HGT attention logits: a[e,h] = <q[dst],kt[src]>_h * prel[h] / sqrt(DIM)
// plus atomicMax into segment max. q stride = 384 (view into kqv buffer).
// ---------------------------------------------------------------------------
__global__ void attn_logits_kernel(const float* __restrict__ kt,
                                   const float* __restrict__ q,
                                   const int* __restrict__ src,
                                   const int* __restrict__ dst, int E,
                                   int dstOff, const float* __restrict__ prel,
                                   float* __restrict__ alphaOut,
                                   unsigned int* __restrict__ mEnc) {
    int t = blockIdx.x * blockDim.x + threadIdx.x;
    if (t >= E * HEADS) return;
    const int e = t >> 2, h = t & 3;
    const int s = src[e], d = dst[e];
    const float* kv = kt + (size_t)s * HID + h * DIM;
    const float* qv = q + (size_t)d * KQV + h * DIM;
    float a = 0.f;
#pragma unroll
    for (int i = 0; i < DIM; ++i) a = fmaf(qv[i], kv[i], a);
    a *= prel[h] * 0.17677669529663687f;  // 1/sqrt(32)
    alphaOut[(size_t)e * HEADS + h] = a;
    atomicMax(mEnc + (size_t)(d + dstOff) * HEADS + h, encF(a));
}

// ---------------------------------------------------------------------------
// HGT aggregation: w = exp(a - m[dst]); sSum[dst,h] += w;
// agg[dst,h,:] += w * vt[src,h,:].  One wave per edge, lane = d.
// ---------------------------------------------------------------------------
__global__ void attn_agg_kernel(const float* __restrict__ vt,
                                const int* __restrict__ src,
                                const int* __restrict__ dst, int E, int dstOff,
                                const float* __restrict__ alpha,
                                const unsigned int* __restrict__ mEnc,
                                float* __restrict__ sSum,
                                float* __restrict__ agg) {
    const int wave = (blockIdx.x * blockDim.x + threadIdx.x) >> 5;
    const int lane = threadIdx.x & 31;
    if (wave >= E) return;
    const int s = src[wave];
    const int gd = dst[wave] + dstOff;
#pragma unroll
    for (int h = 0; h < HEADS; ++h) {
        const float a = alpha[(size_t)wave * HEADS + h];
        const float m = decF(mEnc[(size_t)gd * HEADS + h]);
        const float w = expf(a - m);
        if (lane == 0) atomicAdd(sSum + (size_t)gd * HEADS + h, w);
        const float mv = vt[(size_t)s * HID + h * DIM + lane];
        atomicAdd(agg + (size_t)gd * HID + h * DIM + lane, w * mv);
    }
}

// normalize by segment sum then GELU, in place
__global__ void norm_gelu_kernel(float* __restrict__ agg,
                                 const float* __restrict__ sSum, int N) {
    int t = blockIdx.x * blockDim.x + threadIdx.x;
    if (t >= N * HID) return;
    const int n = t >> 7;
    const int h = (t & 127) >> 5;
    const float denom = sSum[(size_t)n * HEADS + h] + 1e-16f;
    agg[t] = gelu_exact(agg[t] / denom);
}

__global__ void gelu_kernel(float* __restrict__ x, int n) {
    int t = blockIdx.x * blockDim.x + threadIdx.x;
    if (t < n) x[t] = gelu_exact(x[t]);
}

// LayerNorm over 128 features, one block (128 thr) per row, in place
__global__ void layernorm_kernel(float* __restrict__ x,
                                 const float* __restrict__ g,
                                 const float* __restrict__ b) {
    __shared__ float red[HID];
    const int row = blockIdx.x;
    const int c = threadIdx.x;
    const float v = x[(size_t)row * HID + c];
    red[c] = v;
    __syncthreads();
    for (int s = 64; s > 0; s >>= 1) {
        if (c < s) red[c] += red[c + s];
        __syncthreads();
    }
    const float mu = red[0] * (1.0f / HID);
    __syncthreads();
    const float d = v - mu;
    red[c] = d * d;
    __syncthreads();
    for (int s = 64; s > 0; s >>= 1) {
        if (c < s) red[c] += red[c + s];
        __syncthreads();
    }
    const float var = red[0] * (1.0f / HID);
    x[(size_t)row * HID + c] = d * rsqrtf(var + 1e-5f) * g[c] + b[c];
}

// ---------------------------------------------------------------------------
// GAT scorer
// ---------------------------------------------------------------------------
__global__ void gat_x1_kernel(const float* __restrict__ h,
                              const float* __restrict__ Wg,
                              float* __restrict__ x1, int N) {
    int n = blockIdx.x * blockDim.x + threadIdx.x;
    if (n >= N) return;
    float a = 0.f;
#pragma unroll 8
    for (int i = 0; i < HID; ++i) a = fmaf(h[(size_t)n * HID + i], Wg[i], a);
    x1[n] = a;
}

// srcP==nullptr -> self loops (src=dst=e)
__global__ void gat_logits_kernel(const float* __restrict__ x1,
                                  const int* __restrict__ srcP,
                                  const int* __restrict__ dstP, int E,
                                  int srcOff, int dstOff,
                                  const float* __restrict__ attS,
                                  const float* __restrict__ attD,
                                  float* __restrict__ eOut,
                                  unsigned int* __restrict__ mEnc) {
    int e = blockIdx.x * blockDim.x + threadIdx.x;
    if (e >= E) return;
    const int s = srcP ? srcP[e] + srcOff : e;
    const int d = dstP ? dstP[e] + dstOff : e;
    float v = x1[s] * attS[0] + x1[d] * attD[0];
    v = (v > 0.f) ? v : 0.2f * v;  // leaky_relu 0.2
    eOut[e] = v;
    atomicMax(mEnc + d, encF(v));
}

__global__ void gat_accum_kernel(const float* __restrict__ x1,
                                 const int* __restrict__ srcP,
                                 const int* __restrict__ dstP, int E,
                                 int srcOff, int dstOff,
                                 const float* __restrict__ eIn,
                                 const unsigned int* __restrict__ mEnc,
                                 float* __restrict__ sSum,
                                 float* __restrict__ num) {
    int e = blockIdx.x * blockDim.x + threadIdx.x;
    if (e >= E) return;
    const int s = srcP ? srcP[e] + srcOff : e;
    const int d = dstP ? dstP[e] + dstOff : e;
    const float w = expf(eIn[e] - decF(mEnc[d]));
    atomicAdd(sSum + d, w);
    atomicAdd(num + d, w * x1[s]);
}

__global__ void gat_score_kernel(const float* __restrict__ num,
                                 const float* __restrict__ s1,
                                 const float* __restrict__ bgat,
                                 float* __restrict__ score, int N) {
    int n = blockIdx.x * blockDim.x + threadIdx.x;
    if (n >= N) return;
    score[n] = num[n] / (s1[n] + 1e-16f) + bgat[0];
}

// ---------------------------------------------------------------------------
// Top-K (K=8) + gather hp = h[perm] * tanh(vals). Single block, 256 threads.
// ---------------------------------------------------------------------------
__global__ void topk_kernel(const float* __restrict__ score, int N,
                            const float* __restrict__ h,
                            float* __restrict__ hp) {
    __shared__ float bv[256];
    __shared__ int bi[256];
    __shared__ int chosenI[TOPK];
    __shared__ float chosenV[TOPK];
    const int t = threadIdx.x;
    for (int k = 0; k < TOPK; ++k) {
        float best = -INFINITY;
        int besti = N;
        for (int i = t; i < N; i += 256) {
            bool taken = false;
            for (int j = 0; j < k; ++j) taken |= (chosenI[j] == i);
            if (taken) continue;
            const float v = score[i];
            if (v > best || (v == best && i < besti)) {
                best = v;
                besti = i;
            }
        }
        bv[t] = best;
        bi[t] = besti;
        __syncthreads();
        for (int s = 128; s > 0; s >>= 1) {
            if (t < s) {
                if (bv[t + s] > bv[t] ||
                    (bv[t + s] == bv[t] && bi[t + s] < bi[t])) {
                    bv[t] = bv[t + s];
                    bi[t] = bi[t + s];
                }
            }
            __syncthreads();
        }
        if (t == 0) {
            chosenI[k] = bi[0];
            chosenV[k] = bv[0];
        }
        __syncthreads();
    }
    for (int idx = t; idx < TOPK * HID; idx += 256) {
        const int r = idx >> 7, c = idx & 127;
        hp[idx] = h[(size_t)chosenI[r] * HID + c] * tanhf(chosenV[r]);
    }
}

// ---------------------------------------------------------------------------
// Final MLP 1024->128->64->16 (gelu each), nan_to_num. Single block, 128 thr.
// ---------------------------------------------------------------------------
__global__ void mlp_kernel(const float* __restrict__ hp,
                           const float* __restrict__ W1,
                           const float* __restrict__ b1,
                           const float* __restrict__ W2,
                           const float* __restrict__ b2,
                           const float* __restrict__ W3,
                           const float* __restrict__ b3,
                           float* __restrict__ out) {
    __shared__ float h1[128];
    __shared__ float h2[64];
    const int t = threadIdx.x;
    {
        float a = b1[t];
        for (int i = 0; i < TOPK * HID; ++i) a = fmaf(hp[i], W1[i * 128 + t], a);
        h1[t] = gelu_exact(a);
    }
    __syncthreads();
    if (t < 64) {
        float a = b2[t];
        for (int i = 0; i < 128; ++i) a = fmaf(h1[i], W2[i * 64 + t], a);
        h2[t] = gelu_exact(a);
    }
    __syncthreads();
    if (t < 16) {
        float a = b3[t];
        for (int i = 0; i < 64; ++i) a = fmaf(h2[i], W3[i * 16 + t], a);
        float v = gelu_exact(a);
        if (isnan(v)) v = 0.f;
        if (isinf(v))
            v = (v > 0.f) ? 3.4028234663852886e38f : -3.4028234663852886e38f;
        out[t] = v;
    }
}

// ---------------------------------------------------------------------------
// Host orchestration
// ---------------------------------------------------------------------------
extern "C" void kernel_launch(void* const* d_in, const int* in_sizes, int n_in,
                              void* d_out, int out_size, void* d_ws,
                              size_t ws_size, hipStream_t stream) {
    (void)n_in;
    (void)out_size;
    const float* x_a = (const float*)d_in[0];
    const float* x_b = (const float*)d_in[1];
    const int* ei[3] = {(const int*)d_in[2], (const int*)d_in[3],
                        (const int*)d_in[4]};
    const float* Wkqv = (const float*)d_in[5];
    const float* bkqv = (const float*)d_in[6];
    const float* Wout = (const float*)d_in[7];
    const float* bout = (const float*)d_in[8];
    const float* skip = (const float*)d_in[9];
    const float* arel = (const float*)d_in[10];
    const float* mrel = (const float*)d_in[11];
    const float* prel = (const float*)d_in[12];
    const float* ln_g = (const float*)d_in[13];
    const float* ln_b = (const float*)d_in[14];
    const float* Wgat = (const float*)d_in[15];
    const float* att_src = (const float*)d_in[16];
    const float* att_dst = (const float*)d_in[17];
    const float* bgat = (const float*)d_in[18];
    const float* W1 = (const float*)d_in[19];
    const float* b1 = (const float*)d_in[20];
    const float* W2 = (const float*)d_in[21];
    const float* b2 = (const float*)d_in[22];
    const float* W3 = (const float*)d_in[23];
    const float* b3 = (const float*)d_in[24];

    const int Na = in_sizes[0] / HID;   // 20000
    const int Nb = in_sizes[1] / HID;   // 20000
    const int N = Na + Nb;              // 40000
    const int E = in_sizes[2] / 2;      // 150000 per edge type

    // ---- carve workspace ----
    char* p = (char*)d_ws;
    char* pend = p + ws_size;
    auto carve = [&](size_t bytes) -> void* {
        void* r = (void*)p;
        p += (bytes + 255) & ~(size_t)255;
        return r;
    };
    float* h0 = (float*)carve((size_t)N * HID * 4);
    float* h1 = (float*)carve((size_t)N * HID * 4);
    float* kqvA = (float*)carve((size_t)Na * KQV * 4);
    float* kqvB = (float*)carve((size_t)Nb * KQV * 4);
    float* ktBuf = (float*)carve((size_t)Na * HID * 4);  // reused per edge type
    float* vtB[3];
    for (int i = 0; i < 3; ++i) vtB[i] = (float*)carve((size_t)Na * HID * 4);
    float* alphaBuf = (float*)carve((size_t)3 * E * HEADS * 4);
    unsigned int* mEnc = (unsigned int*)carve((size_t)N * HEADS * 4);
    float* sSum = (float*)carve((size_t)N * HEADS * 4);
    float* agg = (float*)carve((size_t)N * HID * 4);
    float* x1 = (float*)carve((size_t)N * 4);
    float* eGat = (float*)carve(((size_t)3 * E + N) * 4);
    unsigned int* m1 = (unsigned int*)carve((size_t)N * 4);
    float* s1 = (float*)carve((size_t)N * 4);
    float* numB = (float*)carve((size_t)N * 4);
    float* score = (float*)carve((size_t)N * 4);
    float* hp = (float*)carve((size_t)TOPK * HID * 4);
    if (p > pend) return;  // workspace too small — bail

    // NT=4 when N is a multiple of 64 (kqv 384, out-proj 128), NT=2 for N=32.
    auto gemm = [&](const float* A, int lda, const float* B, int ldb,
                    const float* bias, float* C, int ldc, int M, int Ncols,
                    int K, int ep, const float* skipX, int ldskip,
                    const float* gate) {
        if ((Ncols & 63) == 0) {
            const int waves = (M >> 4) * (Ncols >> 6);
            const int blocks = (waves * 32 + 127) / 128;
            wmma_gemm_f32_kernel<4><<<blocks, 128, 0, stream>>>(
                A, lda, B, ldb, bias, C, ldc, M, Ncols, K, ep, skipX, ldskip,
                gate);
        } else {
            const int waves = (M >> 4) * (Ncols >> 5);
            const int blocks = (waves * 32 + 127) / 128;
            wmma_gemm_f32_kernel<2><<<blocks, 128, 0, stream>>>(
                A, lda, B, ldb, bias, C, ldc, M, Ncols, K, ep, skipX, ldskip,
                gate);
        }
    };

    // h0 := [x_a ; x_b]
    hipMemcpyAsync(h0, x_a, (size_t)Na * HID * 4, hipMemcpyDeviceToDevice,
                   stream);
    hipMemcpyAsync(h0 + (size_t)Na * HID, x_b, (size_t)Nb * HID * 4,
                   hipMemcpyDeviceToDevice, stream);

    float* cur = h0;
    float* nxt = h1;
    for (int L = 0; L < 3; ++L) {
        float* curA = cur;
        float* curB = cur + (size_t)Na * HID;
        float* nxtA = nxt;
        float* nxtB = nxt + (size_t)Na * HID;

        fill_u32_kernel<<<(N * HEADS + 255) / 256, 256, 0, stream>>>(
            mEnc, ENC_NEG_INF, N * HEADS);
        fill_f32_kernel<<<(N * HEADS + 255) / 256, 256, 0, stream>>>(
            sSum, 0.f, N * HEADS);
        fill_f32_kernel<<<(N * HID + 255) / 256, 256, 0, stream>>>(agg, 0.f,
                                                                   N * HID);

        // kqv projections (WMMA)
        gemm(curA, HID, Wkqv + (size_t)(L * 2 + 0) * HID * KQV, KQV,
             bkqv + (size_t)(L * 2 + 0) * KQV, kqvA, KQV, Na, KQV, HID, 0,
             nullptr, 0, nullptr);
        gemm(curB, HID, Wkqv + (size_t)(L * 2 + 1) * HID * KQV, KQV,
             bkqv + (size_t)(L * 2 + 1) * KQV, kqvB, KQV, Nb, KQV, HID, 0,
             nullptr, 0, nullptr);

        for (int et = 0; et < 3; ++et) {
            const float* kqvS = (et < 2) ? kqvA : kqvB;  // source type
            const float* kqvD = (et == 1) ? kqvB : kqvA; // dest type (q)
            const int dstOff = (et == 1) ? Na : 0;
            // per-head relation transforms (WMMA): kt, vt
            for (int h = 0; h < HEADS; ++h) {
                gemm(kqvS + h * DIM, KQV,
                     arel + (size_t)((L * 3 + et) * HEADS + h) * DIM * DIM, DIM,
                     nullptr, ktBuf + h * DIM, HID, Na, DIM, DIM, 0, nullptr, 0,
                     nullptr);
                gemm(kqvS + 2 * HID + h * DIM, KQV,
                     mrel + (size_t)((L * 3 + et) * HEADS + h) * DIM * DIM, DIM,
                     nullptr, vtB[et] + h * DIM, HID, Na, DIM, DIM, 0, nullptr,
                     0, nullptr);
            }
            // attention logits + segment max (kt consumed here -> ktBuf reused)
            attn_logits_kernel<<<(E * HEADS + 255) / 256, 256, 0, stream>>>(
                ktBuf, kqvD + HID, ei[et], ei[et] + E, E, dstOff,
                prel + (size_t)(L * 3 + et) * HEADS,
                alphaBuf + (size_t)et * E * HEADS, mEnc);
        }
        // exp / segment-sum / weighted aggregation
        for (int et = 0; et < 3; ++et) {
            const int dstOff = (et == 1) ? Na : 0;
            const int blocks = (E * 32 + 127) / 128;  // one wave per edge
            attn_agg_kernel<<<blocks, 128, 0, stream>>>(
                vtB[et], ei[et], ei[et] + E, E, dstOff,
                alphaBuf + (size_t)et * E * HEADS, mEnc, sSum, agg);
        }
        // normalize + gelu (in place)
        norm_gelu_kernel<<<(N * HID + 255) / 256, 256, 0, stream>>>(agg, sSum,
                                                                    N);
        // output projection + gated skip (WMMA)
        gemm(agg, HID, Wout + (size_t)(L * 2 + 0) * HID * HID, HID,
             bout + (size_t)(L * 2 + 0) * HID, nxtA, HID, Na, HID, HID, 1, curA,
             HID, skip + (L * 2 + 0));
        gemm(agg + (size_t)Na * HID, HID,
             Wout + (size_t)(L * 2 + 1) * HID * HID, HID,
             bout + (size_t)(L * 2 + 1) * HID, nxtB, HID, Nb, HID, HID, 1, curB,
             HID, skip + (L * 2 + 1));
        if (L == 0) {
            layernorm_kernel<<<Na, HID, 0, stream>>>(nxtA, ln_g, ln_b);
            layernorm_kernel<<<Nb, HID, 0, stream>>>(nxtB, ln_g + HID,
                                                     ln_b + HID);
        }
        gelu_kernel<<<(N * HID + 255) / 256, 256, 0, stream>>>(nxt, N * HID);
        float* tmp = cur;
        cur = nxt;
        nxt = tmp;
    }

    // ---- GAT scorer over homogeneous graph (h = cur, already [ha;hb]) ----
    const float* h = cur;
    gat_x1_kernel<<<(N + 255) / 256, 256, 0, stream>>>(h, Wgat, x1, N);
    fill_u32_kernel<<<(N + 255) / 256, 256, 0, stream>>>(m1, ENC_NEG_INF, N);
    fill_f32_kernel<<<(N + 255) / 256, 256, 0, stream>>>(s1, 0.f, N);
    fill_f32_kernel<<<(N + 255) / 256, 256, 0, stream>>>(numB, 0.f, N);

    // edge groups: (src,dst,srcOff,dstOff): aa(0,0), ab(0,+Na), ba(+Na,0), self
    const int gb = (E + 255) / 256;
    gat_logits_kernel<<<gb, 256, 0, stream>>>(x1, ei[0], ei[0] + E, E, 0, 0,
                                              att_src, att_dst, eGat, m1);
    gat_logits_kernel<<<gb, 256, 0, stream>>>(x1, ei[1], ei[1] + E, E, 0, Na,
                                              att_src, att_dst, eGat + E, m1);
    gat_logits_kernel<<<gb, 256, 0, stream>>>(x1, ei[2], ei[2] + E, E, Na, 0,
                                              att_src, att_dst, eGat + 2 * E,
                                              m1);
    gat_logits_kernel<<<(N + 255) / 256, 256, 0, stream>>>(
        x1, nullptr, nullptr, N, 0, 0, att_src, att_dst, eGat + 3 * E, m1);

    gat_accum_kernel<<<gb, 256, 0, stream>>>(x1, ei[0], ei[0] + E, E, 0, 0,
                                             eGat, m1, s1, numB);
    gat_accum_kernel<<<gb, 256, 0, stream>>>(x1, ei[1], ei[1] + E, E, 0, Na,
                                             eGat + E, m1, s1, numB);
    gat_accum_kernel<<<gb, 256, 0, stream>>>(x1, ei[2], ei[2] + E, E, Na, 0,
                                             eGat + 2 * E, m1, s1, numB);
    gat_accum_kernel<<<(N + 255) / 256, 256, 0, stream>>>(
        x1, nullptr, nullptr, N, 0, 0, eGat + 3 * E, m1, s1, numB);

    gat_score_kernel<<<(N + 255) / 256, 256, 0, stream>>>(numB, s1, bgat, score,
                                                          N);

    // ---- top-K + pooled features + MLP ----
    topk_kernel<<<1, 256, 0, stream>>>(score, N, h, hp);
    mlp_kernel<<<1, 128, 0, stream>>>(hp, W1, b1, W2, b2, W3, b3,
                                      (float*)d_out);
}